// PacketTimeLSTM_1_31190052504107
// MI455X (gfx1250) — compile-verified
//
#include <hip/hip_runtime.h>
#include <hip/hip_bf16.h>

#define NF   10000     // features
#define HH   64        // hidden
#define NC   2
#define GATES 256      // 4H
#define KW   65        // H+1
#define WPB  4         // waves per block (wave32)
#define BLK1 (WPB * 32)

typedef float v2f __attribute__((ext_vector_type(2)));
typedef float v8f __attribute__((ext_vector_type(8)));

__device__ __forceinline__ float sigmoidf(float x) { return 1.0f / (1.0f + __expf(-x)); }

// ws layout (floats): [0..63] C_sum, [64..127] Hm_sum, [128] cnt, [129] anyMask
__global__ void k0_init(const int* __restrict__ mask, float* __restrict__ ws) {
  __shared__ int red[256];
  const int t = threadIdx.x;
  int s = 0;
  for (int i = t; i < NF; i += 256) s += (mask[i] == 1) ? 1 : 0;
  red[t] = s;
  __syncthreads();
  for (int off = 128; off > 0; off >>= 1) {
    if (t < off) red[t] += red[t + off];
    __syncthreads();
  }
  if (t < 128) ws[t] = 0.0f;
  if (t == 0) {
    const int total = red[0];
    ws[128] = fmaxf((float)total, 1.0f);
    ws[129] = (total > 0) ? 1.0f : 0.0f;
  }
}

__global__ void __launch_bounds__(BLK1)
k1_cell(const int* __restrict__ timp,
        const float* __restrict__ X,
        const int* __restrict__ mask,
        const int* __restrict__ last_occ,
        const float* __restrict__ Ht,
        const float* __restrict__ c_t,
        const float* __restrict__ W,        // [F, 256, 65]
        const float* __restrict__ bias,     // [F, 256]
        const float* __restrict__ xT_w,     // [F, 64]
        const float* __restrict__ xT_b,     // [F, 64]
        const float* __restrict__ delT_w,   // [F, 128]
        const float* __restrict__ cinp_w,   // [F, 64]
        const float* __restrict__ cf_w,     // [F, 64]
        const float* __restrict__ cout_w,   // [F, 64]
        float* __restrict__ outH,           // [F, 64]  (= d_out + 2)
        float* __restrict__ ws) {
  __shared__ float sh_ht[WPB][HH];
  __shared__ float sh_g [WPB][GATES];
  __shared__ float sh_c [WPB][HH];
  __shared__ float sh_hm[WPB][HH];

  const int lane = threadIdx.x & 31;
  const int wv   = threadIdx.x >> 5;
  const int f    = blockIdx.x * WPB + wv;          // 2500 * 4 == 10000, no tail
  const bool hi  = lane >= 16;
  const int mrow = lane & 15;
  const int koff = hi ? 2 : 0;

  const float Xf    = X[f];
  const float delta = (float)timp[0] - (float)last_occ[f];
  const float mf    = (mask[f] == 1) ? 1.0f : 0.0f;
  const float anyM  = ws[129];

  // stage Ht[f,:] into LDS (2 contiguous values per lane)
  {
    const float2 hv = ((const float2*)(Ht + (size_t)f * HH))[lane];
    sh_ht[wv][2 * lane]     = hv.x;
    sh_ht[wv][2 * lane + 1] = hv.y;
  }
  __syncthreads();

  // ---- gates[g] = sum_k W[f,g,1+k] * Ht[f,k]  via V_WMMA_F32_16X16X4_F32 ----
  // A tile: 16 gate-rows x 4 k;  B: input slice broadcast across all 16 columns,
  // so every column of D equals the matvec for those 16 rows.
  const float* Wf = W + (size_t)f * GATES * KW;
  for (int rt = 0; rt < 16; ++rt) {
    v8f acc = {0.f, 0.f, 0.f, 0.f, 0.f, 0.f, 0.f, 0.f};
    const float* Arow = Wf + (size_t)(rt * 16 + mrow) * KW + 1 + koff;
    if (rt < 15)  // pull next row tile toward the caches
      __builtin_prefetch(Wf + (size_t)((rt + 1) * 16 + mrow) * KW, 0, 0);
#pragma unroll
    for (int kt = 0; kt < 16; ++kt) {
      v2f a, b;
      a.x = Arow[4 * kt];                 // A[m, k0+koff]
      a.y = Arow[4 * kt + 1];             // A[m, k0+koff+1]
      b.x = sh_ht[wv][4 * kt + koff];     // B[k0+koff, n]  (broadcast over n)
      b.y = sh_ht[wv][4 * kt + koff + 1];
      acc = __builtin_amdgcn_wmma_f32_16x16x4_f32(false, a, false, b,
                                                  (short)0, acc, false, false);
    }
    // D layout: lane 0 VGPR j -> M=j ; lane 16 VGPR j -> M=8+j (column N=0)
    if (lane == 0 || lane == 16) {
      float* gdst = &sh_g[wv][rt * 16 + (hi ? 8 : 0)];
#pragma unroll
      for (int j = 0; j < 8; ++j) gdst[j] = acc[j];
    }
  }
  __syncthreads();

  // ---- elementwise gate math: 2 h-values per lane ----
#pragma unroll
  for (int t = 0; t < 2; ++t) {
    const int h = lane + 32 * t;
    const size_t fh = (size_t)f * HH + h;
    const float ct = c_t[h];
    // add bias + the k=0 (X) column that was excluded from the WMMA sweep
    const float gi_pre = sh_g[wv][h]        + bias[(size_t)f * GATES + h]        + Wf[(size_t)h * KW] * Xf;
    const float gf_pre = sh_g[wv][64 + h]   + bias[(size_t)f * GATES + 64 + h]   + Wf[(size_t)(64 + h) * KW] * Xf;
    const float go_pre = sh_g[wv][128 + h]  + bias[(size_t)f * GATES + 128 + h]  + Wf[(size_t)(128 + h) * KW] * Xf;
    const float pc_pre = sh_g[wv][192 + h]  + bias[(size_t)f * GATES + 192 + h]  + Wf[(size_t)(192 + h) * KW] * Xf;

    const float gi  = sigmoidf(gi_pre + cinp_w[fh] * ct);
    const float gf  = sigmoidf(gf_pre + cf_w[fh] * ct);
    const float Tt  = sigmoidf(xT_w[fh] * Xf + xT_b[fh] + delT_w[(size_t)f * 2 * HH + h] * delta);
    const float dTo = delT_w[(size_t)f * 2 * HH + HH + h] * delta;
    const float pc  = tanhf(pc_pre);
    const float agg = gf * ct + gi * Tt * pc;
    const float go  = sigmoidf(go_pre + cout_w[fh] * agg + dTo);
    const float hn  = go * tanhf(agg);

    outH[fh] = (anyM > 0.5f) ? (mf * hn) : Ht[fh];   // H_curr
    sh_c [wv][h] = mf * agg;                         // -> C_curr numerator
    sh_hm[wv][h] = mf * hn;                          // -> H_mean numerator
  }
  __syncthreads();

  // ---- block reduction (4 features) then one atomic set per block ----
  const int t128 = threadIdx.x;
  if (t128 < 128) {
    const int sel = t128 >> 6;   // 0 -> C_sum, 1 -> Hm_sum
    const int idx = t128 & 63;
    float s = 0.f;
#pragma unroll
    for (int w = 0; w < WPB; ++w) s += sel ? sh_hm[w][idx] : sh_c[w][idx];
    atomicAdd(ws + t128, s);
  }
}

__global__ void k2_mlp(const float* __restrict__ w1, const float* __restrict__ b1,
                       const float* __restrict__ w2, const float* __restrict__ b2,
                       const float* __restrict__ ws, float* __restrict__ out) {
  __shared__ float inp[2 * HH];
  __shared__ float hid[2 * HH];
  __shared__ float logit[NC];
  const int t = threadIdx.x;           // 128 threads
  const float cnt = ws[128];
  inp[t] = ws[t] / cnt;                // [C_curr | H_mean]
  __syncthreads();
  float acc = b1[t];
  for (int j = 0; j < 2 * HH; ++j) acc += w1[t * 2 * HH + j] * inp[j];
  hid[t] = fmaxf(acc, 0.0f);
  __syncthreads();
  if (t < NC) {
    float a = b2[t];
    for (int j = 0; j < 2 * HH; ++j) a += w2[t * 2 * HH + j] * hid[j];
    logit[t] = a;
  }
  __syncthreads();
  if (t == 0) {
    const float m  = fmaxf(logit[0], logit[1]);
    const float e0 = __expf(logit[0] - m), e1 = __expf(logit[1] - m);
    out[0] = e0 / (e0 + e1);
    out[1] = e1 / (e0 + e1);
  }
  if (t < HH) out[2 + (size_t)NF * HH + t] = ws[t] / cnt;  // C_curr
}

extern "C" void kernel_launch(void* const* d_in, const int* in_sizes, int n_in,
                              void* d_out, int out_size, void* d_ws, size_t ws_size,
                              hipStream_t stream) {
  (void)in_sizes; (void)n_in; (void)out_size; (void)ws_size;
  const int*   timp   = (const int*)  d_in[0];
  const float* X      = (const float*)d_in[1];
  /* d_in[2] X_hap unused by reference */
  const int*   mask   = (const int*)  d_in[3];
  const int*   last   = (const int*)  d_in[4];
  const float* Ht     = (const float*)d_in[5];
  /* d_in[6] Ct unused by reference */
  const float* c_t    = (const float*)d_in[7];
  const float* W      = (const float*)d_in[8];
  const float* bias   = (const float*)d_in[9];
  const float* xT_w   = (const float*)d_in[10];
  const float* xT_b   = (const float*)d_in[11];
  const float* delT_w = (const float*)d_in[12];
  const float* cinp_w = (const float*)d_in[13];
  const float* cf_w   = (const float*)d_in[14];
  const float* cout_w = (const float*)d_in[15];
  const float* w1     = (const float*)d_in[16];
  const float* b1     = (const float*)d_in[17];
  const float* w2     = (const float*)d_in[18];
  const float* b2     = (const float*)d_in[19];
  float* out = (float*)d_out;
  float* ws  = (float*)d_ws;

  hipLaunchKernelGGL(k0_init, dim3(1), dim3(256), 0, stream, mask, ws);
  hipLaunchKernelGGL(k1_cell, dim3(NF / WPB), dim3(BLK1), 0, stream,
                     timp, X, mask, last, Ht, c_t, W, bias,
                     xT_w, xT_b, delT_w, cinp_w, cf_w, cout_w,
                     out + 2, ws);
  hipLaunchKernelGGL(k2_mlp, dim3(1), dim3(128), 0, stream, w1, b1, w2, b2, ws, out);
}